// Attention_62543313764936
// MI455X (gfx1250) — compile-verified
//
#include <hip/hip_runtime.h>

typedef __attribute__((ext_vector_type(2))) float v2f;
typedef __attribute__((ext_vector_type(8))) float v8f;

#define DM 512
#define NQKV 1536
#define TOKENS 32768   // B*C*D*W = 4*16*16*32
#define LDSTRIDE 516   // 512 + 4 pad -> conflict-free 16x4 A-fragment reads

static __device__ __forceinline__ v8f wmma4(v2f a, v2f b, v8f c) {
  // V_WMMA_F32_16X16X4_F32 : D = A(16x4) * B(4x16) + C(16x16)
  return __builtin_amdgcn_wmma_f32_16x16x4_f32(
      /*neg_a=*/false, a, /*neg_b=*/false, b,
      /*c_mod=*/(short)0, c, /*reuse_a=*/false, /*reuse_b=*/false);
}

// LDS byte offset of a generic pointer into __shared__ memory.
// ISA 10.2: in the LDS aperture the low 32 bits of the flat address are the LDS offset.
static __device__ __forceinline__ unsigned ldsOff(const void* p) {
  return (unsigned)(unsigned long long)p;
}

// ASYNCcnt-tracked global->LDS 16B copy (CDNA5 ISA).
static __device__ __forceinline__ void async_g2l_b128(unsigned loff, const void* gaddr) {
  asm volatile("global_load_async_to_lds_b128 %0, %1, off"
               :: "v"(loff), "v"(gaddr) : "memory");
}
static __device__ __forceinline__ void wait_async0() {
  asm volatile("s_wait_asynccnt 0x0" ::: "memory");
}

// ---------------------------------------------------------------------------
// Kernel 0: 16x16-tiled transpose  dst[N][K] = src[K][N]
// grid: (rows/16)*(cols/16) blocks, block: 256 threads
// ---------------------------------------------------------------------------
__global__ __launch_bounds__(256) void transpose_kernel(
    const float* __restrict__ src, float* __restrict__ dst, int rows, int cols) {
  __shared__ float tl[16][17];
  const int nbx = cols >> 4;
  const int bx = blockIdx.x % nbx;
  const int by = blockIdx.x / nbx;
  const int tx = threadIdx.x & 15;
  const int ty = threadIdx.x >> 4;
  tl[ty][tx] = src[(size_t)(by * 16 + ty) * cols + bx * 16 + tx];
  __syncthreads();
  dst[(size_t)(bx * 16 + ty) * rows + by * 16 + tx] = tl[tx][ty];
}

// ---------------------------------------------------------------------------
// Kernel 1: fused LayerNorm + QKV projection
//   x tile async-staged to LDS -> LN -> xn (padded LDS) -> qkv = xn @ Wt^T
// grid: 2048 blocks (16 tokens), block: 256 threads (8 waves)
// ---------------------------------------------------------------------------
__global__ __launch_bounds__(256) void ln_qkv_kernel(
    const float* __restrict__ x, const float* __restrict__ gamma,
    const float* __restrict__ beta, const float* __restrict__ WtQ,  // [1536][512]
    float* __restrict__ qkv) {
  __shared__ float xraw[16 * 512];     // 32 KB, async-staged
  __shared__ float xn[16 * LDSTRIDE];  // 33 KB, normalized + padded

  const int blk = blockIdx.x;
  const int t = threadIdx.x;

  // ---- async stage x tile: 2048 x 16B chunks over 256 threads ----
  const float* xg = x + (size_t)blk * 16 * DM;
#pragma unroll
  for (int i = 0; i < 8; ++i) {
    const unsigned idx = (unsigned)(t + i * 256) * 4;
    async_g2l_b128(ldsOff(&xraw[idx]), xg + idx);
  }
  wait_async0();
  __syncthreads();

  // ---- LayerNorm: 16 threads per row, 32 elems each ----
  const int row = t >> 4;
  const int sub = t & 15;
  float vals[32];
  float s1 = 0.f, s2 = 0.f;
#pragma unroll
  for (int j = 0; j < 32; ++j) {
    float v = xraw[row * 512 + sub + 16 * j];
    vals[j] = v; s1 += v; s2 += v * v;
  }
#pragma unroll
  for (int o = 8; o >= 1; o >>= 1) {   // reduce within 16-lane groups (wave32)
    s1 += __shfl_xor(s1, o, 16);
    s2 += __shfl_xor(s2, o, 16);
  }
  const float mean = s1 * (1.0f / 512.0f);
  const float var  = s2 * (1.0f / 512.0f) - mean * mean;
  const float rstd = rsqrtf(var + 1e-5f);
#pragma unroll
  for (int j = 0; j < 32; ++j) {
    int col = sub + 16 * j;
    xn[row * LDSTRIDE + col] = (vals[j] - mean) * rstd * gamma[col] + beta[col];
  }
  __syncthreads();

  // ---- GEMM: each wave owns 12 N-tiles, in ILP pairs; B rows stream b64 ----
  const int wave = t >> 5;
  const int lane = t & 31;
  const int lrow = lane & 15;
  const int hi   = lane >> 4;
  const int kh   = hi * 2;

  for (int p = 0; p < 6; ++p) {
    const int nt0 = wave + 8 * (2 * p);
    const int nt1 = wave + 8 * (2 * p + 1);
    v8f acc0 = {}, acc1 = {};
    const float* wp0 = WtQ + (size_t)(nt0 * 16 + lrow) * DM;  // lane's own row
    const float* wp1 = WtQ + (size_t)(nt1 * 16 + lrow) * DM;
    __builtin_prefetch(wp0, 0, 3);
    __builtin_prefetch(wp1, 0, 3);
    __builtin_prefetch(wp0 + 256, 0, 3);
    __builtin_prefetch(wp1 + 256, 0, 3);
    for (int kb = 0; kb < 512; kb += 4) {
      v2f a  = *(const v2f*)&xn[lrow * LDSTRIDE + kb + kh];
      v2f b0 = *(const v2f*)&wp0[kb + kh];
      v2f b1 = *(const v2f*)&wp1[kb + kh];
      acc0 = wmma4(a, b0, acc0);
      acc1 = wmma4(a, b1, acc1);
    }
    float* o0 = qkv + (size_t)(blk * 16 + 8 * hi) * NQKV + nt0 * 16 + lrow;
    float* o1 = qkv + (size_t)(blk * 16 + 8 * hi) * NQKV + nt1 * 16 + lrow;
#pragma unroll
    for (int r = 0; r < 8; ++r) {
      o0[(size_t)r * NQKV] = acc0[r];
      o1[(size_t)r * NQKV] = acc1[r];
    }
  }
}

// ---------------------------------------------------------------------------
// Kernel 2: per-(b,h,c,d) attention over W=32, head_dim=64, fully LDS-fed.
// qkv indexed with the torch-.view strides; vhat written at new-view flat
// offsets so the result is already [B,C,D,W,512] contiguous.
// grid: 8192 blocks, block: 32 threads (one wave)
// ---------------------------------------------------------------------------
#define QSTR 196 // 192 + 4 pad: conflict-free, 16B-aligned rows
#define SSTR 34  // 32 + 2 pad: b64-aligned + conflict-free softmax rows
__global__ __launch_bounds__(32) void attn_kernel(
    const float* __restrict__ qkv, float* __restrict__ vhat) {
  __shared__ float qt[32 * QSTR];  // staged q|k|v rows (24.5 KB)
  __shared__ float sc[32 * SSTR];  // scores (4.3 KB)

  const int blk = blockIdx.x;
  const int d = blk & 15, c = (blk >> 4) & 15, h = (blk >> 8) & 7, b = blk >> 11;
  const float* base = qkv + (size_t)b * 12582912 + (size_t)h * 1572864 +
                      (size_t)c * 98304 + (size_t)d * 6144;
  // token w at base + w*192 ; q = +e, k = +64+e, v = +128+e (e in [0,64))

  const int lane = threadIdx.x;
  const int lrow = lane & 15;
  const int hi   = lane >> 4;

  // ---- async stage the 32x192 qkv block: 1536 x 16B chunks, 48 per lane ----
#pragma unroll 8
  for (int i = 0; i < 48; ++i) {
    const int ci = lane + i * 32;
    const int w  = ci / 48;
    const int cc = (ci - w * 48) * 4;
    async_g2l_b128(ldsOff(&qt[w * QSTR + cc]), base + (size_t)w * 192 + cc);
  }
  wait_async0();
  __syncthreads();

  // ---- scores = q @ k^T (M=32,N=32,K=64) ----
  v8f s00 = {}, s01 = {}, s10 = {}, s11 = {};
  for (int kb = 0; kb < 64; kb += 4) {
    const int e = kb + hi * 2;
    v2f a0 = *(const v2f*)&qt[lrow * QSTR + e];
    v2f a1 = *(const v2f*)&qt[(lrow + 16) * QSTR + e];
    v2f b0 = *(const v2f*)&qt[lrow * QSTR + 64 + e];
    v2f b1 = *(const v2f*)&qt[(lrow + 16) * QSTR + 64 + e];
    s00 = wmma4(a0, b0, s00);
    s01 = wmma4(a0, b1, s01);
    s10 = wmma4(a1, b0, s10);
    s11 = wmma4(a1, b1, s11);
  }
  const float scale = 8.0f;  // HEAD_DIM ** 0.5 (reference multiplies!)
#pragma unroll
  for (int r = 0; r < 8; ++r) {
    sc[(r + 8 * hi) * SSTR + lrow]           = s00[r] * scale;
    sc[(r + 8 * hi) * SSTR + 16 + lrow]      = s01[r] * scale;
    sc[(16 + r + 8 * hi) * SSTR + lrow]      = s10[r] * scale;
    sc[(16 + r + 8 * hi) * SSTR + 16 + lrow] = s11[r] * scale;
  }
  __syncthreads();

  // ---- softmax: lane == query row ----
  float rowv[32];
  float m = -3.402823466e38f;
#pragma unroll
  for (int j = 0; j < 32; ++j) { rowv[j] = sc[lane * SSTR + j]; m = fmaxf(m, rowv[j]); }
  float sum = 0.f;
#pragma unroll
  for (int j = 0; j < 32; ++j) { rowv[j] = __expf(rowv[j] - m); sum += rowv[j]; }
  const float inv = 1.0f / sum;
#pragma unroll
  for (int j = 0; j < 32; ++j) sc[lane * SSTR + j] = rowv[j] * inv;
  __syncthreads();

  // ---- vhat = att @ v (M=32,N=64,K=32) ----
  v8f o[2][4];
#pragma unroll
  for (int i = 0; i < 2; ++i)
#pragma unroll
    for (int j = 0; j < 4; ++j) o[i][j] = (v8f)0.0f;

  for (int kb = 0; kb < 32; kb += 4) {
    const int kk = kb + hi * 2;
    v2f a0 = *(const v2f*)&sc[lrow * SSTR + kk];
    v2f a1 = *(const v2f*)&sc[(lrow + 16) * SSTR + kk];
#pragma unroll
    for (int nt = 0; nt < 4; ++nt) {
      const int e = nt * 16 + lrow;
      v2f bv;
      bv.x = qt[kk * QSTR + 128 + e];
      bv.y = qt[(kk + 1) * QSTR + 128 + e];
      o[0][nt] = wmma4(a0, bv, o[0][nt]);
      o[1][nt] = wmma4(a1, bv, o[1][nt]);
    }
  }

  float* vb = vhat + (size_t)b * 4194304 + (size_t)h * 524288 +
              (size_t)c * 32768 + (size_t)d * 2048;
#pragma unroll
  for (int mi = 0; mi < 2; ++mi)
#pragma unroll
    for (int nt = 0; nt < 4; ++nt)
#pragma unroll
      for (int r = 0; r < 8; ++r) {
        int w = mi * 16 + r + 8 * hi;
        vb[(size_t)w * 64 + nt * 16 + lrow] = o[mi][nt][r];
      }
}

// ---------------------------------------------------------------------------
// Kernel 3: out = vhat[32768,512] @ W_out[512,512] + b_out  (B from Wt_out)
// grid: 2048 blocks, block: 256 threads (8 waves, 4 N-tiles each)
// ---------------------------------------------------------------------------
__global__ __launch_bounds__(256) void out_proj_kernel(
    const float* __restrict__ vh, const float* __restrict__ WtO,  // [512][512]
    const float* __restrict__ bout, float* __restrict__ out) {
  __shared__ float tile[16 * LDSTRIDE];

  const int blk = blockIdx.x;
  const int t = threadIdx.x;

  // ---- async stage vhat tile directly into padded LDS layout ----
  const float* vg = vh + (size_t)blk * 16 * DM;
#pragma unroll
  for (int i = 0; i < 2; ++i) {
    const int ci = t + i * 256;            // 0..511 chunk id
    const int r  = ci >> 5;                // row 0..15
    const int cc = (ci & 31) * 4;          // col (floats)
    async_g2l_b128(ldsOff(&tile[r * LDSTRIDE + cc]), vg + (size_t)r * DM + cc);
  }
  wait_async0();
  __syncthreads();

  const int wave = t >> 5;
  const int lane = t & 31;
  const int lrow = lane & 15;
  const int hi   = lane >> 4;
  const int kh   = hi * 2;

  for (int p = 0; p < 2; ++p) {
    const int nt0 = wave + 8 * (2 * p);
    const int nt1 = wave + 8 * (2 * p + 1);
    v8f acc0 = {}, acc1 = {};
    const float* wp0 = WtO + (size_t)(nt0 * 16 + lrow) * DM;
    const float* wp1 = WtO + (size_t)(nt1 * 16 + lrow) * DM;
    __builtin_prefetch(wp0, 0, 3);
    __builtin_prefetch(wp1, 0, 3);
    __builtin_prefetch(wp0 + 256, 0, 3);
    __builtin_prefetch(wp1 + 256, 0, 3);
    for (int kb = 0; kb < 512; kb += 4) {
      v2f a  = *(const v2f*)&tile[lrow * LDSTRIDE + kb + kh];
      v2f b0 = *(const v2f*)&wp0[kb + kh];
      v2f b1 = *(const v2f*)&wp1[kb + kh];
      acc0 = wmma4(a, b0, acc0);
      acc1 = wmma4(a, b1, acc1);
    }
    const float bias0 = bout[nt0 * 16 + lrow];
    const float bias1 = bout[nt1 * 16 + lrow];
    float* o0 = out + (size_t)(blk * 16 + 8 * hi) * DM + nt0 * 16 + lrow;
    float* o1 = out + (size_t)(blk * 16 + 8 * hi) * DM + nt1 * 16 + lrow;
#pragma unroll
    for (int r = 0; r < 8; ++r) {
      o0[(size_t)r * DM] = acc0[r] + bias0;
      o1[(size_t)r * DM] = acc1[r] + bias1;
    }
  }
}

extern "C" void kernel_launch(void* const* d_in, const int* in_sizes, int n_in,
                              void* d_out, int out_size, void* d_ws, size_t ws_size,
                              hipStream_t stream) {
  (void)in_sizes; (void)n_in; (void)out_size; (void)ws_size;
  const float* x     = (const float*)d_in[0];
  const float* gamma = (const float*)d_in[1];
  const float* beta  = (const float*)d_in[2];
  const float* Wqkv  = (const float*)d_in[3];
  const float* Wout  = (const float*)d_in[4];
  const float* bout  = (const float*)d_in[5];
  float* out  = (float*)d_out;

  float* qkv  = (float*)d_ws;                       // 32768*1536 f32 = 192 MB
  float* vhat = qkv + (size_t)TOKENS * NQKV;        // 32768*512  f32 =  64 MB
  float* WtQ  = vhat + (size_t)TOKENS * DM;         // 1536*512   f32 =   3 MB
  float* WtO  = WtQ + (size_t)NQKV * DM;            // 512*512    f32 =   1 MB

  transpose_kernel<<<(DM / 16) * (NQKV / 16), 256, 0, stream>>>(Wqkv, WtQ, DM, NQKV);
  transpose_kernel<<<(DM / 16) * (DM / 16), 256, 0, stream>>>(Wout, WtO, DM, DM);
  ln_qkv_kernel<<<TOKENS / 16, 256, 0, stream>>>(x, gamma, beta, WtQ, qkv);
  attn_kernel<<<8192, 32, 0, stream>>>(qkv, vhat);
  out_proj_kernel<<<TOKENS / 16, 256, 0, stream>>>(vhat, WtO, bout, out);
}